// PlenoxelModel_52759378264705
// MI455X (gfx1250) — compile-verified
//
#include <hip/hip_runtime.h>

#define GRES 160
#define NSAMP 128
#define NEAR_T 0.1f
#define FAR_T 10.0f

typedef __attribute__((ext_vector_type(2))) float v2f;
typedef __attribute__((ext_vector_type(8))) float v8f;

__global__ __launch_bounds__(128) void plenoxel_fused_kernel(
    const float* __restrict__ ro, const float* __restrict__ rd,
    const float* __restrict__ density, const float* __restrict__ sh,
    float* __restrict__ out, int R)
{
    // LDS staging: 384 rows of (9 SH coeffs zero-padded to 12) for WMMA A tiles
    __shared__ float sh_lds[384 * 12];     // 18432 B
    __shared__ float pre_lds[384];         // projected colors pre-sigmoid
    __shared__ float scan_lds[NSAMP];      // transmittance product scan
    __shared__ float basis_lds[12];        // SH basis (padded), uniform per ray
    __shared__ float wred[4 * 5];          // cross-wave reduction

    const int ray  = blockIdx.x;
    const int tid  = threadIdx.x;          // == sample index along the ray
    const int lane = tid & 31;
    const int wv   = tid >> 5;

    // ---- uniform per-ray data (uniform addresses -> scalarized) ----
    const float ox = ro[ray * 3 + 0], oy = ro[ray * 3 + 1], oz = ro[ray * 3 + 2];
    const float rx = rd[ray * 3 + 0], ry = rd[ray * 3 + 1], rz = rd[ray * 3 + 2];
    const float dnorm = sqrtf(rx * rx + ry * ry + rz * rz);

    // ---- SH degree-2 basis (constant over the ray) ----
    {
        const float inv = 1.0f / (dnorm + 1e-8f);
        const float x = rx * inv, y = ry * inv, z = rz * inv;
        if (tid == 0) {
            basis_lds[0]  =  0.28209479177387814f;
            basis_lds[1]  = -0.48860251190291987f * y;
            basis_lds[2]  =  0.48860251190291987f * z;
            basis_lds[3]  = -0.48860251190291987f * x;
            basis_lds[4]  =  1.0925484305920792f * x * y;
            basis_lds[5]  = -1.0925484305920792f * y * z;
            basis_lds[6]  =  0.31539156525252005f * (2.0f * z * z - x * x - y * y);
            basis_lds[7]  = -1.0925484305920792f * x * z;
            basis_lds[8]  =  0.5462742152960396f * (x * x - y * y);
            basis_lds[9]  = 0.0f; basis_lds[10] = 0.0f; basis_lds[11] = 0.0f;
        }
    }

    // ---- sample position ----
    const float tval = NEAR_T + (float)tid * ((FAR_T - NEAR_T) / (float)(NSAMP - 1));
    const float px = ox + tval * rx;
    const float py = oy + tval * ry;
    const float pz = oz + tval * rz;

    // world -> voxel: (p - (-1)) / 2 * 160, clipped to [0, 159]
    float vx = (px + 1.0f) * (0.5f * (float)GRES);
    float vy = (py + 1.0f) * (0.5f * (float)GRES);
    float vz = (pz + 1.0f) * (0.5f * (float)GRES);
    vx = fminf(fmaxf(vx, 0.0f), (float)(GRES - 1));
    vy = fminf(fmaxf(vy, 0.0f), (float)(GRES - 1));
    vz = fminf(fmaxf(vz, 0.0f), (float)(GRES - 1));

    const float fx = floorf(vx), fy = floorf(vy), fz = floorf(vz);
    const float dx = vx - fx, dy = vy - fy, dz = vz - fz;
    const int x0 = (int)fx, y0 = (int)fy, z0 = (int)fz;
    const int x1 = min(x0 + 1, GRES - 1);
    const int y1 = min(y0 + 1, GRES - 1);
    const int z1 = min(z0 + 1, GRES - 1);

    const int   xs[2] = { x0, x1 };
    const int   ys[2] = { y0, y1 };
    const int   zs[2] = { z0, z1 };
    const float wxa[2] = { 1.0f - dx, dx };
    const float wya[2] = { 1.0f - dy, dy };
    const float wza[2] = { 1.0f - dz, dz };

    // ---- 8-corner trilinear gather (the memory-bound core) ----
    float dens = 0.0f;
    float shv[27];
#pragma unroll
    for (int k = 0; k < 27; ++k) shv[k] = 0.0f;

#pragma unroll
    for (int cz = 0; cz < 2; ++cz) {
#pragma unroll
        for (int cy = 0; cy < 2; ++cy) {
#pragma unroll
            for (int cx = 0; cx < 2; ++cx) {
                const float w = wza[cz] * wya[cy] * wxa[cx];
                const int vidx = (zs[cz] * GRES + ys[cy]) * GRES + xs[cx];
                dens = fmaf(w, density[vidx], dens);
                const float* __restrict__ sp = sh + (size_t)vidx * 27u;
#pragma unroll
                for (int k = 0; k < 27; ++k) shv[k] = fmaf(w, sp[k], shv[k]);
            }
        }
    }

    // ---- stage SH rows for WMMA: row = sample*3 + channel, 12 floats (9 + pad) ----
#pragma unroll
    for (int c = 0; c < 3; ++c) {
        const int rb = (tid * 3 + c) * 12;
#pragma unroll
        for (int k = 0; k < 9; ++k) sh_lds[rb + k] = shv[c * 9 + k];
        sh_lds[rb + 9] = 0.0f; sh_lds[rb + 10] = 0.0f; sh_lds[rb + 11] = 0.0f;
    }
    __syncthreads();

    // ---- SH projection via V_WMMA_F32_16X16X4_F32 with broadcast-B ----
    // Each wave handles its own 96 rows (samples 32*wv .. 32*wv+31, 3 channels)
    // as 6 tiles of 16 rows; K = 12 (padded) in 3 chunks of 4.
    // A layout (16x4 f32): lane l holds row M=l&15; VGPR0/1 = K = k0+2*(l>>4), +1.
    // B layout (4x16 f32): lane l holds col N=l&15; VGPR0/1 = K = k0+2*(l>>4), +1.
    // B[k][n] = basis[k] for all n  =>  every column of D = dot(A_row, basis).
    const int half2 = (lane >> 4) << 1;
    const int mrow  = lane & 15;
#pragma unroll
    for (int tile = 0; tile < 6; ++tile) {
        const int m0  = wv * 96 + tile * 16;
        const int row = m0 + mrow;
        v8f acc = {};
#pragma unroll
        for (int k0 = 0; k0 < 12; k0 += 4) {
            const int kk = k0 + half2;
            const v2f a = *(const v2f*)&sh_lds[row * 12 + kk];   // ds_load_b64, conflict-free
            const v2f b = *(const v2f*)&basis_lds[kk];           // ds_load_b64 broadcast
            acc = __builtin_amdgcn_wmma_f32_16x16x4_f32(
                false, a, false, b, (short)0, acc, false, false);
        }
        // D layout: VGPR r -> (row m0+r, col=lane) for lanes 0-15; (row m0+8+r) for 16-31.
        // All columns identical; lanes 0 and 16 write the 16 row results.
        if (mrow == 0) {
            const int rb = m0 + ((lane >> 4) << 3);
#pragma unroll
            for (int r = 0; r < 8; ++r) pre_lds[rb + r] = acc[r];
        }
    }
    __syncthreads();

    // ---- sigmoid + alpha ----
    float cr = pre_lds[tid * 3 + 0];
    float cg = pre_lds[tid * 3 + 1];
    float cb = pre_lds[tid * 3 + 2];
    cr = 1.0f / (1.0f + expf(-cr));
    cg = 1.0f / (1.0f + expf(-cg));
    cb = 1.0f / (1.0f + expf(-cb));

    float dist = (tid == NSAMP - 1) ? 1e10f : ((FAR_T - NEAR_T) / (float)(NSAMP - 1));
    dist *= dnorm;
    const float alpha = 1.0f - expf(-fmaxf(dens, 0.0f) * dist);
    const float f = 1.0f - alpha + 1e-10f;

    // ---- inclusive product scan (Hillis-Steele) over 128 samples ----
    scan_lds[tid] = f;
    __syncthreads();
    float run = f;
#pragma unroll
    for (int off = 1; off < NSAMP; off <<= 1) {
        const float xv = (tid >= off) ? scan_lds[tid - off] : 1.0f;
        __syncthreads();
        run *= xv;
        scan_lds[tid] = run;
        __syncthreads();
    }
    const float T   = (tid == 0) ? 1.0f : scan_lds[tid - 1];  // exclusive transmittance
    const float wgt = alpha * T;

    // ---- reductions: rgb (3), depth, acc ----
    float vals[5] = { wgt * cr, wgt * cg, wgt * cb, wgt * tval, wgt };
#pragma unroll
    for (int q = 0; q < 5; ++q) {
        float v = vals[q];
#pragma unroll
        for (int off = 16; off > 0; off >>= 1) v += __shfl_down(v, off, 32);
        if (lane == 0) wred[wv * 5 + q] = v;
    }
    __syncthreads();
    if (tid < 5) {
        const float s = wred[tid] + wred[5 + tid] + wred[10 + tid] + wred[15 + tid];
        if (tid < 3)        out[ray * 3 + tid] = s;      // rgb
        else if (tid == 3)  out[R * 3 + ray]   = s;      // depth
        else                out[R * 4 + ray]   = s;      // acc
    }
}

extern "C" void kernel_launch(void* const* d_in, const int* in_sizes, int n_in,
                              void* d_out, int out_size, void* d_ws, size_t ws_size,
                              hipStream_t stream) {
    const float* ro      = (const float*)d_in[0];  // [R,3]
    const float* rd      = (const float*)d_in[1];  // [R,3]
    const float* density = (const float*)d_in[2];  // [160^3]
    const float* sh      = (const float*)d_in[3];  // [160^3 * 27]
    float* out = (float*)d_out;                    // rgb[R*3] | depth[R] | acc[R]
    const int R = in_sizes[0] / 3;

    plenoxel_fused_kernel<<<R, NSAMP, 0, stream>>>(ro, rd, density, sh, out, R);
}